// MultiTaskGNN_v6_78795470012791
// MI455X (gfx1250) — compile-verified
//
#include <hip/hip_runtime.h>

#define N_NODES   100000
#define N_EDGES   1600000
#define IN_DIM    64
#define HID       128
#define HID2      256
#define N_GRAPHS  512
#define OUT_DIM   3
#define BN_EPS    1e-5f

// padded LDS strides (floats): +4 rotates banks by 4 per row -> conflict-free
#define XS 68    // IN_DIM + 4
#define ZS 132   // HID + 4
#define TS 260   // HID2 + 4

typedef __attribute__((ext_vector_type(2))) float v2f;
typedef __attribute__((ext_vector_type(8))) float v8f;
typedef __attribute__((ext_vector_type(4))) int   v4i;

typedef __attribute__((address_space(1))) v4i* glb_v4i_ptr;
typedef __attribute__((address_space(3))) v4i* lds_v4i_ptr;

__device__ __forceinline__ v8f wmma4(v2f a, v2f b, v8f c) {
  // D = A(16x4 f32) * B(4x16 f32) + C(16x16 f32)
  return __builtin_amdgcn_wmma_f32_16x16x4_f32(
      /*neg_a=*/false, a, /*neg_b=*/false, b,
      /*c_mod=*/(short)0, c, /*reuse_a=*/false, /*reuse_b=*/false);
}

// ---- async global->LDS (CDNA5 GLOBAL_LOAD_ASYNC_TO_LDS_B128, ASYNCcnt) ----
#if defined(__has_builtin)
#  if __has_builtin(__builtin_amdgcn_global_load_async_to_lds_b128)
#    define HAS_ASYNC_LDS 1
#  endif
#endif
#ifndef HAS_ASYNC_LDS
#  define HAS_ASYNC_LDS 0
#endif

__device__ __forceinline__ void async_copy16(const float* g, float* l) {
#if HAS_ASYNC_LDS
  __builtin_amdgcn_global_load_async_to_lds_b128(
      (glb_v4i_ptr)g, (lds_v4i_ptr)l, /*imm offset=*/0, /*cpol=*/0);
#else
  *(float4*)l = *(const float4*)g;
#endif
}

__device__ __forceinline__ void async_join() {
#if HAS_ASYNC_LDS
#  if defined(__has_builtin) && __has_builtin(__builtin_amdgcn_s_wait_asynccnt)
  __builtin_amdgcn_s_wait_asynccnt(0);
#  else
  asm volatile("s_wait_asynccnt 0" ::: "memory");
#  endif
#endif
}

// ---------------------------------------------------------------- zero
__global__ __launch_bounds__(256) void zero_f4(float4* __restrict__ p, int n4) {
  int i = blockIdx.x * 256 + threadIdx.x;
  if (i < n4) p[i] = make_float4(0.f, 0.f, 0.f, 0.f);
}

// ---------------------------------------------------------------- h = x @ W0 + b0
__global__ __launch_bounds__(256) void gemm0_kernel(
    const float* __restrict__ x, const float* __restrict__ W0,
    const float* __restrict__ b0, float* __restrict__ h) {
  __shared__ float xs[16 * XS];
  const int tid = threadIdx.x;
  const int node0 = blockIdx.x * 16;

  // stage 16x64 input rows via async DMA to LDS
  {
    int r = tid >> 4, c4 = tid & 15;          // 256 float4 = 16*64 floats
    async_copy16(x + (size_t)(node0 + r) * IN_DIM + c4 * 4, xs + r * XS + c4 * 4);
    async_join();
  }
  __syncthreads();

  const int lane = tid & 31, wv = tid >> 5;
  const int row = lane & 15, kp = lane >> 4;
  const int c = wv * 16 + row;      // 8 waves cover 128 cols

  v8f acc = {};
#pragma unroll 4
  for (int k = 0; k < IN_DIM; k += 4) {
    v2f a, b;
    a.x = xs[row * XS + k + 2 * kp];
    a.y = xs[row * XS + k + 2 * kp + 1];
    b.x = W0[(size_t)(k + 2 * kp) * HID + c];
    b.y = W0[(size_t)(k + 2 * kp + 1) * HID + c];
    acc = wmma4(a, b, acc);
  }
  const float bias = b0[c];
#pragma unroll
  for (int v = 0; v < 8; ++v) {
    int r = v + 8 * kp;
    h[(size_t)(node0 + r) * HID + c] = acc[v] + bias;
  }
}

// ---------------------------------------------------------------- aggr[dst] += h[src]
__global__ __launch_bounds__(256) void scatter_kernel(
    const float* __restrict__ h, const long long* __restrict__ src,
    const long long* __restrict__ dst, float* __restrict__ aggr) {
  int gid = blockIdx.x * 256 + threadIdx.x;   // N_EDGES * 32 threads
  int e = gid >> 5;
  int d0 = (gid & 31) * 4;
  if (e >= N_EDGES) return;
  long long s = src[e], t = dst[e];
  const float4 hv = *(const float4*)(h + (size_t)s * HID + d0);
  float* ap = aggr + (size_t)t * HID + d0;
  atomicAdd(ap + 0, hv.x);
  atomicAdd(ap + 1, hv.y);
  atomicAdd(ap + 2, hv.z);
  atomicAdd(ap + 3, hv.w);
}

// ---------------------------------------------------------------- fused GIN MLP (one layer)
__global__ __launch_bounds__(256) void gin_mlp_kernel(
    float* __restrict__ h, const float* __restrict__ aggr,
    const float* __restrict__ eps, int l,
    const float* __restrict__ W1l, const float* __restrict__ b1l,
    const float* __restrict__ g1l, const float* __restrict__ be1l,
    const float* __restrict__ m1l, const float* __restrict__ v1l,
    const float* __restrict__ W2l, const float* __restrict__ b2l,
    const float* __restrict__ g2l, const float* __restrict__ be2l,
    const float* __restrict__ m2l, const float* __restrict__ v2l) {
  __shared__ float zs[16 * ZS];
  __shared__ float ts[16 * TS];
  const int tid = threadIdx.x;
  const int node0 = blockIdx.x * 16;
  const float ep1 = 1.0f + eps[l];

  // z = (1+eps)*h + aggr for the 16-node tile (fused math -> keep as VGPR copy)
  for (int i = tid; i < 16 * (HID / 4); i += 256) {   // 512 float4
    int r = i >> 5, c4 = i & 31;
    const float4 hv = *(const float4*)(h + (size_t)(node0 + r) * HID + c4 * 4);
    const float4 av = *(const float4*)(aggr + (size_t)(node0 + r) * HID + c4 * 4);
    float4 z;
    z.x = ep1 * hv.x + av.x;  z.y = ep1 * hv.y + av.y;
    z.z = ep1 * hv.z + av.z;  z.w = ep1 * hv.w + av.w;
    *(float4*)(zs + r * ZS + c4 * 4) = z;
  }
  __syncthreads();

  const int lane = tid & 31, wv = tid >> 5;
  const int row = lane & 15, kp = lane >> 4;

  // GEMM1: t = relu(bn1(z @ W1 + b1))   [16 x 256], 2 col-tiles per wave
#pragma unroll
  for (int t = 0; t < 2; ++t) {
    const int c = (wv + t * 8) * 16 + row;
    v8f acc = {};
#pragma unroll 8
    for (int k = 0; k < HID; k += 4) {
      v2f a, b;
      a.x = zs[row * ZS + k + 2 * kp];
      a.y = zs[row * ZS + k + 2 * kp + 1];
      b.x = W1l[(size_t)(k + 2 * kp) * HID2 + c];
      b.y = W1l[(size_t)(k + 2 * kp + 1) * HID2 + c];
      acc = wmma4(a, b, acc);
    }
    const float alpha = g1l[c] * rsqrtf(v1l[c] + BN_EPS);
    const float beta  = (b1l[c] - m1l[c]) * alpha + be1l[c];
#pragma unroll
    for (int v = 0; v < 8; ++v) {
      int r = v + 8 * kp;
      float val = acc[v] * alpha + beta;
      ts[r * TS + c] = fmaxf(val, 0.0f);
    }
  }
  __syncthreads();

  // GEMM2: h = relu(bn2(t @ W2 + b2)) + h   [16 x 128], 1 col-tile per wave
  {
    const int c = wv * 16 + row;
    v8f acc = {};
#pragma unroll 8
    for (int k = 0; k < HID2; k += 4) {
      v2f a, b;
      a.x = ts[row * TS + k + 2 * kp];
      a.y = ts[row * TS + k + 2 * kp + 1];
      b.x = W2l[(size_t)(k + 2 * kp) * HID + c];
      b.y = W2l[(size_t)(k + 2 * kp + 1) * HID + c];
      acc = wmma4(a, b, acc);
    }
    const float alpha = g2l[c] * rsqrtf(v2l[c] + BN_EPS);
    const float beta  = (b2l[c] - m2l[c]) * alpha + be2l[c];
#pragma unroll
    for (int v = 0; v < 8; ++v) {
      int r = v + 8 * kp;
      float* hp = h + (size_t)(node0 + r) * HID + c;
      float res = *hp;
      float val = acc[v] * alpha + beta;
      *hp = fmaxf(val, 0.0f) + res;
    }
  }
}

// ---------------------------------------------------------------- hg[batch[i]] += h[i]
__global__ __launch_bounds__(256) void pool_kernel(
    const float* __restrict__ h, const long long* __restrict__ batch,
    float* __restrict__ hg) {
  int gid = blockIdx.x * 256 + threadIdx.x;   // N_NODES * 32 threads
  int node = gid >> 5;
  int d0 = (gid & 31) * 4;
  if (node >= N_NODES) return;
  long long g = batch[node];
  const float4 hv = *(const float4*)(h + (size_t)node * HID + d0);
  float* p = hg + (size_t)g * HID + d0;
  atomicAdd(p + 0, hv.x);
  atomicAdd(p + 1, hv.y);
  atomicAdd(p + 2, hv.z);
  atomicAdd(p + 3, hv.w);
}

// ---------------------------------------------------------------- o1 = relu(hg @ Wa + ba)
__global__ __launch_bounds__(256) void readout1_kernel(
    const float* __restrict__ hg, const float* __restrict__ Wa,
    const float* __restrict__ ba, float* __restrict__ o1) {
  __shared__ float gs[16 * ZS];
  const int tid = threadIdx.x;
  const int row0 = blockIdx.x * 16;
  for (int i = tid; i < 16 * (HID / 4); i += 256) {   // pure copy -> async DMA
    int r = i >> 5, c4 = i & 31;
    async_copy16(hg + (size_t)(row0 + r) * HID + c4 * 4, gs + r * ZS + c4 * 4);
  }
  async_join();
  __syncthreads();
  const int lane = tid & 31, wv = tid >> 5;
  const int row = lane & 15, kp = lane >> 4;
  const int c = wv * 16 + row;
  v8f acc = {};
#pragma unroll 8
  for (int k = 0; k < HID; k += 4) {
    v2f a, b;
    a.x = gs[row * ZS + k + 2 * kp];
    a.y = gs[row * ZS + k + 2 * kp + 1];
    b.x = Wa[(size_t)(k + 2 * kp) * HID + c];
    b.y = Wa[(size_t)(k + 2 * kp + 1) * HID + c];
    acc = wmma4(a, b, acc);
  }
  const float bias = ba[c];
#pragma unroll
  for (int v = 0; v < 8; ++v) {
    int r = v + 8 * kp;
    o1[(size_t)(row0 + r) * HID + c] = fmaxf(acc[v] + bias, 0.0f);
  }
}

// ---------------------------------------------------------------- o2 = relu(o1 @ Wb + bb), 64 cols
__global__ __launch_bounds__(256) void readout2_kernel(
    const float* __restrict__ o1, const float* __restrict__ Wb,
    const float* __restrict__ bb, float* __restrict__ o2) {
  __shared__ float gs[16 * ZS];
  const int tid = threadIdx.x;
  const int row0 = blockIdx.x * 16;
  for (int i = tid; i < 16 * (HID / 4); i += 256) {   // pure copy -> async DMA
    int r = i >> 5, c4 = i & 31;
    async_copy16(o1 + (size_t)(row0 + r) * HID + c4 * 4, gs + r * ZS + c4 * 4);
  }
  async_join();
  __syncthreads();
  const int lane = tid & 31, wv = tid >> 5;
  if (wv < 4) {                               // 4 col-tiles cover 64 cols
    const int row = lane & 15, kp = lane >> 4;
    const int c = wv * 16 + row;
    v8f acc = {};
#pragma unroll 8
    for (int k = 0; k < HID; k += 4) {
      v2f a, b;
      a.x = gs[row * ZS + k + 2 * kp];
      a.y = gs[row * ZS + k + 2 * kp + 1];
      b.x = Wb[(size_t)(k + 2 * kp) * (HID / 2) + c];
      b.y = Wb[(size_t)(k + 2 * kp + 1) * (HID / 2) + c];
      acc = wmma4(a, b, acc);
    }
    const float bias = bb[c];
#pragma unroll
    for (int v = 0; v < 8; ++v) {
      int r = v + 8 * kp;
      o2[(size_t)(row0 + r) * (HID / 2) + c] = fmaxf(acc[v] + bias, 0.0f);
    }
  }
}

// ---------------------------------------------------------------- out = o2 @ Wc + bc  (64 -> 3)
__global__ __launch_bounds__(256) void readout3_kernel(
    const float* __restrict__ o2, const float* __restrict__ Wc,
    const float* __restrict__ bc, float* __restrict__ out) {
  int gid = blockIdx.x * 256 + threadIdx.x;
  if (gid >= N_GRAPHS * OUT_DIM) return;
  int g = gid / OUT_DIM;
  int j = gid - g * OUT_DIM;
  float s = bc[j];
  const float* op = o2 + (size_t)g * (HID / 2);
#pragma unroll 8
  for (int k = 0; k < HID / 2; ++k) s += op[k] * Wc[k * OUT_DIM + j];
  out[gid] = s;
}

// ---------------------------------------------------------------- host
extern "C" void kernel_launch(void* const* d_in, const int* in_sizes, int n_in,
                              void* d_out, int out_size, void* d_ws, size_t ws_size,
                              hipStream_t stream) {
  const float*     x     = (const float*)d_in[0];
  const long long* ei    = (const long long*)d_in[1];
  const long long* batch = (const long long*)d_in[2];
  const float* W0 = (const float*)d_in[3];
  const float* b0 = (const float*)d_in[4];
  const float* eps = (const float*)d_in[5];
  const float* W1 = (const float*)d_in[6];
  const float* b1 = (const float*)d_in[7];
  const float* g1 = (const float*)d_in[8];
  const float* be1 = (const float*)d_in[9];
  const float* m1 = (const float*)d_in[10];
  const float* v1 = (const float*)d_in[11];
  const float* W2 = (const float*)d_in[12];
  const float* b2 = (const float*)d_in[13];
  const float* g2 = (const float*)d_in[14];
  const float* be2 = (const float*)d_in[15];
  const float* m2 = (const float*)d_in[16];
  const float* v2 = (const float*)d_in[17];
  const float* Wa = (const float*)d_in[18];
  const float* ba = (const float*)d_in[19];
  const float* Wb = (const float*)d_in[20];
  const float* bb = (const float*)d_in[21];
  const float* Wc = (const float*)d_in[22];
  const float* bc = (const float*)d_in[23];
  float* out = (float*)d_out;

  // workspace layout (floats)
  float* h    = (float*)d_ws;
  float* aggr = h    + (size_t)N_NODES * HID;
  float* hg   = aggr + (size_t)N_NODES * HID;
  float* o1   = hg   + (size_t)N_GRAPHS * HID;
  float* o2   = o1   + (size_t)N_GRAPHS * HID;

  const long long src_off = 0, dst_off = N_EDGES;

  gemm0_kernel<<<N_NODES / 16, 256, 0, stream>>>(x, W0, b0, h);

  for (int l = 0; l < 3; ++l) {
    const int n4 = N_NODES * HID / 4;
    zero_f4<<<(n4 + 255) / 256, 256, 0, stream>>>((float4*)aggr, n4);
    scatter_kernel<<<(N_EDGES * 32) / 256, 256, 0, stream>>>(
        h, ei + src_off, ei + dst_off, aggr);
    gin_mlp_kernel<<<N_NODES / 16, 256, 0, stream>>>(
        h, aggr, eps, l,
        W1 + (size_t)l * HID * HID2, b1 + (size_t)l * HID2,
        g1 + (size_t)l * HID2, be1 + (size_t)l * HID2,
        m1 + (size_t)l * HID2, v1 + (size_t)l * HID2,
        W2 + (size_t)l * HID2 * HID, b2 + (size_t)l * HID,
        g2 + (size_t)l * HID, be2 + (size_t)l * HID,
        m2 + (size_t)l * HID, v2 + (size_t)l * HID);
  }

  const int n4g = N_GRAPHS * HID / 4;
  zero_f4<<<(n4g + 255) / 256, 256, 0, stream>>>((float4*)hg, n4g);
  pool_kernel<<<(N_NODES * 32 + 255) / 256, 256, 0, stream>>>(h, batch, hg);
  readout1_kernel<<<N_GRAPHS / 16, 256, 0, stream>>>(hg, Wa, ba, o1);
  readout2_kernel<<<N_GRAPHS / 16, 256, 0, stream>>>(o1, Wb, bb, o2);
  readout3_kernel<<<(N_GRAPHS * OUT_DIM + 255) / 256, 256, 0, stream>>>(o2, Wc, bc, out);

  (void)in_sizes; (void)n_in; (void)out_size; (void)ws_size;
}